// MSSMamba2_80985903333789
// MI455X (gfx1250) — compile-verified
//
#include <hip/hip_runtime.h>
#include <math.h>

// MSSMamba2 forward for gfx1250 (MI455X), fp32 throughout using
// V_WMMA_F32_16X16X4_F32 for every GEMM-shaped stage.
// Workspace requirement: ~63.2M floats (~242 MB).

typedef float v2f __attribute__((ext_vector_type(2)));
typedef float v8f __attribute__((ext_vector_type(8)));

#define L4   4096
#define DM_  256
#define DI_  512
#define DS_  128
#define CD_  768
#define DIP_ 1288
#define NH_  8
#define HD_  64
#define CS_  256
#define NC_  16

__device__ inline v8f wmma4(v2f a, v2f b, v8f c) {
  // D = A(16x4,f32) x B(4x16,f32) + C(16x16,f32)
  return __builtin_amdgcn_wmma_f32_16x16x4_f32(false, a, false, b, (short)0, c,
                                               false, false);
}

__device__ inline float silu_f(float x) { return x / (1.0f + expf(-x)); }

// Token map: row l of direction d's sequence -> flat (h*W+w) index in pts.
// Also doubles as the scatter map from direction token -> cat row.
__device__ inline int tokmap(int d, int l) {
  if (d == 0) return l;
  if (d == 1) return 4095 - l;
  int ll = (d == 2) ? l : (4095 - l);
  return ((ll & 63) << 6) | (ll >> 6); // (ll%64)*64 + ll/64
}

// ---------------------------------------------------------------- generic
// One wave computes a 32x32 tile of C = A(row-major, MxK) * W^T where
// W is row-major NxK. Both fragments are float2 loads.
struct Acc4 { v8f a00, a01, a10, a11; };

__device__ inline Acc4 wave_gemm32(const float* __restrict__ A, int lda,
                                   const float* __restrict__ Bw, int ldb,
                                   int K) {
  int lane = threadIdx.x & 31;
  int lr = lane & 15, lh = lane >> 4;
  v8f z = {};
  Acc4 r; r.a00 = z; r.a01 = z; r.a10 = z; r.a11 = z;
  const float* a0p = A + (size_t)lr * lda + 2 * lh;
  const float* a1p = A + (size_t)(16 + lr) * lda + 2 * lh;
  const float* b0p = Bw + (size_t)lr * ldb + 2 * lh;
  const float* b1p = Bw + (size_t)(16 + lr) * ldb + 2 * lh;
  for (int k = 0; k < K; k += 4) {
    v2f a0 = *(const v2f*)(a0p + k);
    v2f a1 = *(const v2f*)(a1p + k);
    v2f b0 = *(const v2f*)(b0p + k);
    v2f b1 = *(const v2f*)(b1p + k);
    r.a00 = wmma4(a0, b0, r.a00);
    r.a01 = wmma4(a0, b1, r.a01);
    r.a10 = wmma4(a1, b0, r.a10);
    r.a11 = wmma4(a1, b1, r.a11);
  }
  return r;
}

__device__ inline void store_acc(const Acc4& ac, float* __restrict__ C,
                                 int ldc) {
  int lane = threadIdx.x & 31;
  int lr = lane & 15, lh = lane >> 4;
  for (int r = 0; r < 8; ++r) {
    int row = r + 8 * lh;
    C[(size_t)row * ldc + lr]            = ac.a00[r];
    C[(size_t)row * ldc + 16 + lr]       = ac.a01[r];
    C[(size_t)(row + 16) * ldc + lr]     = ac.a10[r];
    C[(size_t)(row + 16) * ldc + 16 + lr] = ac.a11[r];
  }
}

// ---------------------------------------------------------------- kernels
// u4[d][l][c] = pts[c][tokmap(d,l)]
__global__ void k_build_u(const float* __restrict__ pts,
                          float* __restrict__ u4) {
  int idx = blockIdx.x * 256 + threadIdx.x; // 4*4096*256
  int c = idx & 255, l = (idx >> 8) & 4095, d = idx >> 20;
  u4[idx] = pts[(size_t)c * L4 + tokmap(d, l)];
}

// zxb[d] (4096x1280) = u4[d] @ Win[d][0:1280,:]^T
__global__ void k_gemm_inproj(const float* __restrict__ u4,
                              const float* __restrict__ Win,
                              float* __restrict__ zxb) {
  int gw = blockIdx.x * 4 + (threadIdx.x >> 5); // 20480 waves
  int d = gw / 5120, t = gw % 5120;
  int mt = t / 40, nt = t % 40;
  const float* A  = u4 + ((size_t)d * L4 + mt * 32) * DM_;
  const float* Bw = Win + (size_t)d * DIP_ * DM_ + (size_t)(nt * 32) * DM_;
  Acc4 ac = wave_gemm32(A, DM_, Bw, DM_, DM_);
  store_acc(ac, zxb + ((size_t)d * L4 + mt * 32) * 1280 + nt * 32, 1280);
}

// dts[(d*L+l)*8+h] = softplus(u4[d][l] . Win[d][1280+h] + dt_bias)
__global__ void k_dt(const float* __restrict__ u4,
                     const float* __restrict__ Win,
                     const float* __restrict__ dtb,
                     float* __restrict__ dts) {
  int idx = blockIdx.x * 256 + threadIdx.x; // 131072
  int h = idx & 7, l = (idx >> 3) & 4095, d = idx >> 15;
  const float* ur = u4 + ((size_t)d * L4 + l) * DM_;
  const float* wr = Win + (size_t)d * DIP_ * DM_ + (size_t)(1280 + h) * DM_;
  float acc = dtb[d * 8 + h];
  for (int k = 0; k < DM_; ++k) acc += ur[k] * wr[k];
  dts[idx] = (acc > 20.f) ? acc : log1pf(expf(acc));
}

// convo[d][l][j] = silu(conv_b + sum_w xBC[l-3+w][j]*conv_w[j][w])
__global__ void k_conv(const float* __restrict__ zxb,
                       const float* __restrict__ convw,
                       const float* __restrict__ convb,
                       float* __restrict__ convo) {
  int idx = blockIdx.x * 256 + threadIdx.x; // 4*4096*768
  int j = idx % 768;
  int t = idx / 768;
  int l = t & 4095, d = t >> 12;
  float acc = convb[d * 768 + j];
  const float* cw = convw + (size_t)(d * 768 + j) * 4;
  for (int w = 0; w < 4; ++w) {
    int ls = l - 3 + w;
    if (ls >= 0)
      acc += zxb[((size_t)d * L4 + ls) * 1280 + 512 + j] * cw[w];
  }
  convo[idx] = silu_f(acc);
}

// Per (d,chunk,head): inclusive cumsum of dt*A within chunk + chunk decay.
__global__ void k_cumsum(const float* __restrict__ dts,
                         const float* __restrict__ Alog,
                         float* __restrict__ dacsb,
                         float* __restrict__ cdec) {
  int idx = blockIdx.x * 256 + threadIdx.x; // 512
  if (idx >= 512) return;
  int h = idx & 7, c = (idx >> 3) & 15, d = idx >> 7;
  float A = -expf(Alog[d * 8 + h]);
  float run = 0.f;
  for (int i = 0; i < CS_; ++i) {
    size_t o = ((size_t)d * L4 + c * CS_ + i) * 8 + h;
    run += dts[o] * A;
    dacsb[o] = run;
  }
  cdec[(d * 16 + c) * 8 + h] = expf(run);
}

// CB[d][c][i][j] = sum_n C[i][n]*B[j][n]   (256x256, K=128 per chunk)
__global__ void k_cb(const float* __restrict__ convo,
                     float* __restrict__ cbb) {
  int gw = blockIdx.x * 4 + (threadIdx.x >> 5); // 4096 waves
  int dc = gw >> 6, t = gw & 63;
  int mt = t >> 3, nt = t & 7;
  int d = dc >> 4, c = dc & 15;
  const float* A  = convo + ((size_t)d * L4 + c * CS_ + mt * 32) * CD_ + 640;
  const float* Bw = convo + ((size_t)d * L4 + c * CS_ + nt * 32) * CD_ + 512;
  Acc4 ac = wave_gemm32(A, CD_, Bw, CD_, DS_);
  store_acc(ac, cbb + ((size_t)dc * 256 + mt * 32) * 256 + nt * 32, 256);
}

// states[d][c][h][p][n] = sum_s B[s][n] * dt[s]*exp(dacs_last-dacs[s]) * x[s][p]
__global__ void k_states(const float* __restrict__ convo,
                         const float* __restrict__ dts,
                         const float* __restrict__ dacsb,
                         float* __restrict__ states) {
  __shared__ float wsh[CS_];
  int b = blockIdx.x; // (d*16+c)*8+h, 512 blocks, 128 threads
  int h = b & 7, c = (b >> 3) & 15, d = b >> 7;
  int tid = threadIdx.x;
  float dlast = dacsb[((size_t)d * L4 + c * CS_ + 255) * 8 + h];
  for (int i = tid; i < CS_; i += 128) {
    size_t o = ((size_t)d * L4 + c * CS_ + i) * 8 + h;
    wsh[i] = dts[o] * expf(dlast - dacsb[o]);
  }
  __syncthreads();
  int lane = tid & 31, lr = lane & 15, lh = lane >> 4;
  int p0 = (tid >> 5) * 16; // 4 waves, one p-tile each
  const float* xb = convo + ((size_t)d * L4 + c * CS_) * CD_ + h * HD_;
  const float* bb = convo + ((size_t)d * L4 + c * CS_) * CD_ + 512;
  v8f acc[8]; v8f z = {};
  for (int nt = 0; nt < 8; ++nt) acc[nt] = z;
  for (int k = 0; k < CS_; k += 4) {
    int s0 = k + 2 * lh;
    v2f a;
    a[0] = xb[(size_t)s0 * CD_ + p0 + lr] * wsh[s0];
    a[1] = xb[(size_t)(s0 + 1) * CD_ + p0 + lr] * wsh[s0 + 1];
    for (int nt = 0; nt < 8; ++nt) {
      v2f bf;
      bf[0] = bb[(size_t)s0 * CD_ + nt * 16 + lr];
      bf[1] = bb[(size_t)(s0 + 1) * CD_ + nt * 16 + lr];
      acc[nt] = wmma4(a, bf, acc[nt]);
    }
  }
  float* st = states + (size_t)b * HD_ * DS_;
  for (int nt = 0; nt < 8; ++nt)
    for (int r = 0; r < 8; ++r)
      st[(size_t)(p0 + r + 8 * lh) * DS_ + nt * 16 + lr] = acc[nt][r];
}

// Exclusive inter-chunk scan: prev[c]=carry; carry = carry*cdec[c]+states[c]
__global__ void k_scan(const float* __restrict__ states,
                       const float* __restrict__ cdec,
                       float* __restrict__ prevb) {
  int idx = blockIdx.x * 256 + threadIdx.x; // 4*8*64*128 = 262144
  int n = idx & 127, p = (idx >> 7) & 63, h = (idx >> 13) & 7, d = idx >> 16;
  float carry = 0.f;
  for (int c = 0; c < NC_; ++c) {
    size_t si = ((size_t)(((d * 16 + c) * 8 + h) * 64 + p)) * DS_ + n;
    float st = states[si];
    prevb[si] = carry;
    carry = carry * cdec[(d * 16 + c) * 8 + h] + st;
  }
}

// y = y_off + y_diag + D*x  per (d,chunk,head). One wave per i-tile.
__global__ void k_ssd_y(const float* __restrict__ convo,
                        const float* __restrict__ cbb,
                        const float* __restrict__ dacsb,
                        const float* __restrict__ dts,
                        const float* __restrict__ prevb,
                        const float* __restrict__ Dprm,
                        float* __restrict__ ybuf) {
  __shared__ float Xs[CS_ * HD_]; // 64 KB
  __shared__ float dA[CS_];
  __shared__ float dtv[CS_];
  int b = blockIdx.x; // 512 blocks, 512 threads
  int h = b & 7, c = (b >> 3) & 15, d = b >> 7;
  int tid = threadIdx.x;
  for (int i = tid; i < CS_ * HD_; i += 512) {
    int s = i >> 6, p = i & 63;
    Xs[i] = convo[((size_t)d * L4 + c * CS_ + s) * CD_ + h * HD_ + p];
  }
  if (tid < CS_) {
    size_t o = ((size_t)d * L4 + c * CS_ + tid) * 8 + h;
    dA[tid] = dacsb[o];
    dtv[tid] = dts[o];
  }
  __syncthreads();

  int it = tid >> 5; // i-tile 0..15
  int lane = tid & 31, lr = lane & 15, lh = lane >> 4;
  float di = dA[it * 16 + lr];
  float ei = expf(di);
  v8f acc[4]; v8f z = {};
  for (int pt = 0; pt < 4; ++pt) acc[pt] = z;

  // y_off: (exp(dacs_i) * C[i,:]) @ prev[p,:]^T, K = 128 over n
  const float* crow = convo + ((size_t)d * L4 + c * CS_ + it * 16 + lr) * CD_ + 640;
  const float* pv = prevb + (size_t)b * HD_ * DS_;
  for (int k = 0; k < DS_; k += 4) {
    v2f a = *(const v2f*)(crow + k + 2 * lh);
    a *= ei;
    for (int pt = 0; pt < 4; ++pt) {
      v2f bf = *(const v2f*)(pv + (size_t)(pt * 16 + lr) * DS_ + k + 2 * lh);
      acc[pt] = wmma4(a, bf, acc[pt]);
    }
  }

  // y_diag: causal S @ X, j-tiles up to i-tile
  const float* cbrow = cbb + ((size_t)((d * 16 + c) * 256 + it * 16 + lr)) * 256;
  for (int jt = 0; jt <= it; ++jt) {
    for (int kk = 0; kk < 4; ++kk) {
      int jb = jt * 16 + kk * 4 + 2 * lh;
      v2f a;
      for (int v = 0; v < 2; ++v) {
        int j = jb + v;
        int jj = kk * 4 + 2 * lh + v; // local j within tile
        float val = 0.f;
        if (jt < it || jj <= lr)
          val = cbrow[j] * expf(di - dA[j]) * dtv[j];
        a[v] = val;
      }
      for (int pt = 0; pt < 4; ++pt) {
        v2f bf;
        bf[0] = Xs[(size_t)jb * HD_ + pt * 16 + lr];
        bf[1] = Xs[(size_t)(jb + 1) * HD_ + pt * 16 + lr];
        acc[pt] = wmma4(a, bf, acc[pt]);
      }
    }
  }

  float Dp = Dprm[d * 8 + h];
  for (int pt = 0; pt < 4; ++pt)
    for (int r = 0; r < 8; ++r) {
      int i = it * 16 + r + 8 * lh;
      int p = pt * 16 + lr;
      ybuf[((size_t)d * L4 + c * CS_ + i) * DI_ + h * HD_ + p] =
          acc[pt][r] + Dp * Xs[(size_t)i * HD_ + p];
    }
}

// y *= silu(z); RMSNorm over 512 with norm_w. In-place on ybuf.
__global__ void k_gate(const float* __restrict__ zxb,
                       const float* __restrict__ nw,
                       float* __restrict__ ybuf) {
  __shared__ float red[256];
  int row = blockIdx.x; // 16384 rows (d*4096+l)
  int tid = threadIdx.x;
  int d = row >> 12;
  float z0 = zxb[(size_t)row * 1280 + tid];
  float z1 = zxb[(size_t)row * 1280 + 256 + tid];
  float y0 = ybuf[(size_t)row * DI_ + tid];
  float y1 = ybuf[(size_t)row * DI_ + 256 + tid];
  float t0 = y0 * silu_f(z0), t1 = y1 * silu_f(z1);
  red[tid] = t0 * t0 + t1 * t1;
  __syncthreads();
  for (int s = 128; s > 0; s >>= 1) {
    if (tid < s) red[tid] += red[tid + s];
    __syncthreads();
  }
  float sc = rsqrtf(red[0] * (1.0f / 512.0f) + 1e-5f);
  ybuf[(size_t)row * DI_ + tid]       = t0 * sc * nw[d * 512 + tid];
  ybuf[(size_t)row * DI_ + 256 + tid] = t1 * sc * nw[d * 512 + 256 + tid];
}

// out-proj + silu + direction scatter into cat (4096x1024)
__global__ void k_outproj(const float* __restrict__ ybuf,
                          const float* __restrict__ Wout,
                          float* __restrict__ catb) {
  int gw = blockIdx.x * 4 + (threadIdx.x >> 5); // 4096 waves
  int d = gw >> 10, t = gw & 1023;
  int mt = t >> 3, nt = t & 7;
  const float* A  = ybuf + ((size_t)d * L4 + mt * 32) * DI_;
  const float* Bw = Wout + (size_t)d * DM_ * DI_ + (size_t)(nt * 32) * DI_;
  Acc4 ac = wave_gemm32(A, DI_, Bw, DI_, DI_);
  int lane = threadIdx.x & 31, lr = lane & 15, lh = lane >> 4;
  for (int r = 0; r < 8; ++r) {
    int row0 = mt * 32 + r + 8 * lh;
    int row1 = row0 + 16;
    int cr0 = tokmap(d, row0), cr1 = tokmap(d, row1);
    int c0 = d * 256 + nt * 32 + lr, c1 = c0 + 16;
    catb[(size_t)cr0 * 1024 + c0] = silu_f(ac.a00[r]);
    catb[(size_t)cr0 * 1024 + c1] = silu_f(ac.a01[r]);
    catb[(size_t)cr1 * 1024 + c0] = silu_f(ac.a10[r]);
    catb[(size_t)cr1 * 1024 + c1] = silu_f(ac.a11[r]);
  }
}

// final: out(4096x256) = cat(4096x1024) @ final_w^T
__global__ void k_final(const float* __restrict__ catb,
                        const float* __restrict__ Wfin,
                        float* __restrict__ out) {
  int gw = blockIdx.x * 4 + (threadIdx.x >> 5); // 1024 waves
  int mt = gw >> 3, nt = gw & 7;
  Acc4 ac = wave_gemm32(catb + (size_t)(mt * 32) * 1024, 1024,
                        Wfin + (size_t)(nt * 32) * 1024, 1024, 1024);
  store_acc(ac, out + (size_t)(mt * 32) * 256 + nt * 32, 256);
}

// ---------------------------------------------------------------- launch
extern "C" void kernel_launch(void* const* d_in, const int* in_sizes, int n_in,
                              void* d_out, int out_size, void* d_ws,
                              size_t ws_size, hipStream_t stream) {
  (void)in_sizes; (void)n_in; (void)out_size; (void)ws_size;
  const float* pts   = (const float*)d_in[0];
  // d_in[1] (foreground_predictions) unused by reference
  const float* Win   = (const float*)d_in[2];
  const float* convw = (const float*)d_in[3];
  const float* convb = (const float*)d_in[4];
  const float* dtb   = (const float*)d_in[5];
  const float* Alog  = (const float*)d_in[6];
  const float* Dprm  = (const float*)d_in[7];
  const float* nw    = (const float*)d_in[8];
  const float* Wout  = (const float*)d_in[9];
  const float* Wfin  = (const float*)d_in[10];
  float* out = (float*)d_out;

  float* w = (float*)d_ws;
  size_t off = 0;
  float* u4    = w + off; off += (size_t)4 * L4 * DM_;   // 4,194,304
  float* zxb   = w + off; off += (size_t)4 * L4 * 1280;  // 20,971,520
  float* convo = w + off; off += (size_t)4 * L4 * CD_;   // 12,582,912
  float* dts   = w + off; off += (size_t)4 * L4 * NH_;   // 131,072
  float* dacsb = w + off; off += (size_t)4 * L4 * NH_;   // 131,072
  float* cdec  = w + off; off += (size_t)4 * NC_ * NH_;  // 512
  float* cbb   = w + off; off += (size_t)4 * NC_ * CS_ * CS_;       // 4,194,304
  float* stb   = w + off; off += (size_t)4 * NC_ * NH_ * HD_ * DS_; // 4,194,304
  float* pvb   = w + off; off += (size_t)4 * NC_ * NH_ * HD_ * DS_; // 4,194,304
  float* ybuf  = w + off; off += (size_t)4 * L4 * DI_;   // 8,388,608
  float* catb  = w + off; off += (size_t)L4 * 1024;      // 4,194,304
  // total ~63.2M floats (~242 MB)

  k_build_u<<<16384, 256, 0, stream>>>(pts, u4);
  k_gemm_inproj<<<5120, 128, 0, stream>>>(u4, Win, zxb);
  k_dt<<<512, 256, 0, stream>>>(u4, Win, dtb, dts);
  k_conv<<<49152, 256, 0, stream>>>(zxb, convw, convb, convo);
  k_cumsum<<<2, 256, 0, stream>>>(dts, Alog, dacsb, cdec);
  k_cb<<<1024, 128, 0, stream>>>(convo, cbb);
  k_states<<<512, 128, 0, stream>>>(convo, dts, dacsb, stb);
  k_scan<<<1024, 256, 0, stream>>>(stb, cdec, pvb);
  k_ssd_y<<<512, 512, 0, stream>>>(convo, cbb, dacsb, dts, pvb, Dprm, ybuf);
  k_gate<<<16384, 256, 0, stream>>>(zxb, nw, ybuf);
  k_outproj<<<1024, 128, 0, stream>>>(ybuf, Wout, catb);
  k_final<<<256, 128, 0, stream>>>(catb, Wfin, out);
}